// GSSC_83468394431251
// MI455X (gfx1250) — compile-verified
//
#include <hip/hip_runtime.h>
#include <hip/hip_bf16.h>

// ---------------------------------------------------------------------------
// CDNA5 (gfx1250) fp32 WMMA implementation of the low-rank linear-attention
// block. All matrix work uses V_WMMA_F32_16X16X4_F32 (full fp32 precision,
// matches the fp32 reference), wave32 layouts per the CDNA5 ISA:
//   A 16x4 f32 : lanes 0-15 = rows, v0/v1 = K, lanes 16-31 = K+2/K+3
//   B 4x16 f32 : lanes 0-15 = cols, v0/v1 = K, lanes 16-31 = K+2/K+3
//   C/D 16x16  : vgpr j -> M = j + 8*(lane>=16), N = lane&15
//
// Layout choices (set where the intermediates are produced):
//   xh    -> xhT[b][l][n]   (K-major consumer in first_kernel reads v2f)
//   k_pe  -> kpeT[b][rd][n] (same reason)
//   first -> firstT[b][l][rd] (K-major consumer in attn_out reads v2f)
// For xhT/firstT the 8 D-VGPRs of a lane are CONSECUTIVE in the transposed
// index, so the transposed store is a single aligned v8f (2x b128) per lane.
// ---------------------------------------------------------------------------

#define BB   16
#define NN   2048
#define RR   8
#define DPE  32
#define LL   256
#define RD   256             // R*DPE
#define BN   (BB*NN)         // 32768
#define BNL  ((size_t)BN*LL) // 8388608

typedef __attribute__((ext_vector_type(2))) float v2f;
typedef __attribute__((ext_vector_type(8))) float v8f;

static __device__ __forceinline__ v8f wmma4(v2f a, v2f b, v8f c) {
  // D = A(16x4) * B(4x16) + C, fp32 throughout
  return __builtin_amdgcn_wmma_f32_16x16x4_f32(
      /*neg_a=*/false, a, /*neg_b=*/false, b,
      /*c_mod=*/(short)0, c, /*reuse_a=*/false, /*reuse_b=*/false);
}

// ---------------------------------------------------------------------------
// K1: degree-mix + two token projections  xh = xmix@Wx^T, x_rw = xmix@Wrx^T
// grid = BN/16 blocks, 512 threads (16 waves); A tile staged mixed in LDS.
// xh is written TRANSPOSED per batch: xhT[b][l][n].
// ---------------------------------------------------------------------------
__global__ __launch_bounds__(512) void mix_gemm_kernel(
    const float* __restrict__ x, const float* __restrict__ log_deg,
    const float* __restrict__ deg_coef, const float* __restrict__ Wx,
    const float* __restrict__ Wrx, float* __restrict__ xhT,
    float* __restrict__ xrw) {
  __shared__ float amix[16 * LL];
  const int tid = threadIdx.x;
  const int g0 = blockIdx.x * 16;  // token-row base over B*N

  // cooperative stage of the mixed A tile (16 rows x 256)
  for (int i = 0; i < 8; ++i) {
    int idx = i * 512 + tid;
    int row = idx >> 8;
    int col = idx & 255;
    float xv = x[(size_t)(g0 + row) * LL + col];
    float ld = log_deg[g0 + row];
    v2f dc = *(const v2f*)&deg_coef[2 * col];  // [dc0, dc1]
    amix[idx] = xv * (dc.x + ld * dc.y);
  }
  __syncthreads();

  const int w = tid >> 5, t = tid & 31;
  const int lane = t & 15, half = t >> 4;
  const int m0 = w * 16;  // output-feature tile base
  v8f accx = {}; v8f accr = {};
  for (int k = 0; k < LL; k += 4) {
    int kh = k + 2 * half;
    v2f a  = *(const v2f*)&amix[lane * LL + kh];
    v2f bx = *(const v2f*)&Wx[(size_t)(m0 + lane) * LL + kh];
    v2f br = *(const v2f*)&Wrx[(size_t)(m0 + lane) * LL + kh];
    accx = wmma4(a, bx, accx);   // shares A fragment between both GEMMs
    accr = wmma4(a, br, accr);
  }
  // xh transposed store: fixed (lane,half) -> 8 consecutive tokens n
  {
    const int b = g0 >> 11;                       // tiles never cross batch
    const int nb = (g0 & (NN - 1)) + 8 * half;    // multiple of 8 -> 32B align
    const int col = m0 + lane;
    *(v8f*)&xhT[(size_t)b * LL * NN + (size_t)col * NN + nb] = accx;
  }
  for (int j = 0; j < 8; ++j) {
    int row = g0 + j + 8 * half;
    int col = m0 + lane;
    xrw[(size_t)row * LL + col] = accr[j];
  }
}

// ---------------------------------------------------------------------------
// K2: PE projections  q_pe = pe@Wq^T (row layout), k_pe = pe@Wk^T
// k_pe written TRANSPOSED per batch: kpeT[b][rd][n], rd = r*32+d.
// grid = (B*N*R)/128 blocks, 256 threads (8 waves, one 16-row tile each).
// ---------------------------------------------------------------------------
__global__ __launch_bounds__(256) void pe_gemm_kernel(
    const float* __restrict__ pe, const float* __restrict__ Wq,
    const float* __restrict__ Wk, float* __restrict__ qpe,
    float* __restrict__ kpeT) {
  const int tid = threadIdx.x;
  const int w = tid >> 5, t = tid & 31;
  const int lane = t & 15, half = t >> 4;
  const size_t r0 = ((size_t)blockIdx.x * 8 + w) * 16;  // row base over B*N*R
  v8f q0 = {}, q1 = {}, k0v = {}, k1v = {};
  for (int k = 0; k < DPE; k += 4) {
    int kh = k + 2 * half;
    v2f a  = *(const v2f*)&pe[(r0 + lane) * DPE + kh];
    v2f b0 = *(const v2f*)&Wq[(size_t)lane * DPE + kh];
    v2f b1 = *(const v2f*)&Wq[(size_t)(16 + lane) * DPE + kh];
    v2f c0 = *(const v2f*)&Wk[(size_t)lane * DPE + kh];
    v2f c1 = *(const v2f*)&Wk[(size_t)(16 + lane) * DPE + kh];
    q0  = wmma4(a, b0, q0);
    q1  = wmma4(a, b1, q1);
    k0v = wmma4(a, c0, k0v);
    k1v = wmma4(a, c1, k1v);
  }
  // rows r0+8*half .. +7 cover r=0..7 of a single (b,n): r == j below
  const size_t nbflat = (r0 + 8 * half) >> 3;  // flat (b,n)
  const int n = (int)(nbflat & (NN - 1));
  const int b = (int)(nbflat >> 11);
  const size_t tb = (size_t)b * RD * NN;
  for (int j = 0; j < 8; ++j) {
    size_t row = r0 + j + 8 * half;  // flat (b,n,r), r=j
    qpe[row * DPE + lane]      = q0[j];
    qpe[row * DPE + 16 + lane] = q1[j];
    kpeT[tb + (size_t)(j * DPE + lane) * NN + n]      = k0v[j];
    kpeT[tb + (size_t)(j * DPE + 16 + lane) * NN + n] = k1v[j];
  }
}

// ---------------------------------------------------------------------------
// K3: scale[b,n] = sum_r pe_r^T (Wrpe^T Wrpe2) pe_r   (quadratic-form trick)
// grid = BN/8 blocks, 256 threads (8 waves, wave-per-token).
// ---------------------------------------------------------------------------
__global__ __launch_bounds__(256) void scale_kernel(
    const float* __restrict__ pe, const float* __restrict__ Wrpe,
    const float* __restrict__ Wrpe2, float* __restrict__ scale) {
  __shared__ float Msh[DPE * DPE];
  const int tid = threadIdx.x;
  for (int i = tid; i < DPE * DPE; i += 256) {
    int d = i >> 5, dp = i & 31;
    float s = 0.f;
    for (int e = 0; e < DPE; ++e)
      s += Wrpe[e * DPE + d] * Wrpe2[e * DPE + dp];
    Msh[i] = s;  // Msh[d][d'] = (Wrpe^T Wrpe2)[d][d']
  }
  __syncthreads();
  const int w = tid >> 5, t = tid & 31;
  const size_t tok = (size_t)blockIdx.x * 8 + w;
  float acc = 0.f;
  for (int r = 0; r < RR; ++r) {
    const float* per = &pe[(tok * RR + r) * DPE];
    float pv = per[t];
    float y = 0.f;
    for (int d = 0; d < DPE; ++d) y += per[d] * Msh[d * DPE + t];
    acc += y * pv;
  }
  for (int off = 16; off > 0; off >>= 1) acc += __shfl_xor(acc, off, 32);
  if (t == 0) scale[tok] = acc;
}

// ---------------------------------------------------------------------------
// K4: first[b, rd, l] = sum_n k_pe[b,n,rd] * xh[b,n,l]    (K = N = 2048)
// Both operands are K-major and stored [feature][n] -> contiguous v2f loads;
// each lane streams linearly through one 8KB row. Output stored TRANSPOSED
// firstT[b][l][rd] as one aligned v8f per lane.
// grid = (16 rd-tiles, 16 batches), 512 threads (16 waves over l-tiles).
// ---------------------------------------------------------------------------
__global__ __launch_bounds__(512) void first_kernel(
    const float* __restrict__ kpeT, const float* __restrict__ xhT,
    float* __restrict__ firstT) {
  const int tid = threadIdx.x;
  const int w = tid >> 5, t = tid & 31;
  const int lane = t & 15, half = t >> 4;
  const int rd0 = blockIdx.x * 16;
  const int b   = blockIdx.y;
  const int l0  = w * 16;
  const size_t tb = (size_t)b * RD * NN;  // per-batch stride for both inputs
  const float* aRow = &kpeT[tb + (size_t)(rd0 + lane) * NN];
  const float* bRow = &xhT[tb + (size_t)(l0 + lane) * NN];
  v8f acc = {};
  for (int n = 0; n < NN; n += 4) {
    int kh = n + 2 * half;
    v2f a  = *(const v2f*)&aRow[kh];
    v2f bb = *(const v2f*)&bRow[kh];
    acc = wmma4(a, bb, acc);
  }
  // lane's 8 D values are consecutive in rd -> single v8f store
  *(v8f*)&firstT[(size_t)b * RD * LL + (size_t)(l0 + lane) * RD + rd0 +
                 8 * half] = acc;
}

// ---------------------------------------------------------------------------
// K5: fused  att = q_pe@first  ->  out = att@Wout^T + bout + scale*x_rw
// grid = BN/16 blocks, 512 threads; att tile held in LDS between phases.
// ---------------------------------------------------------------------------
__global__ __launch_bounds__(512) void attn_out_kernel(
    const float* __restrict__ qpe, const float* __restrict__ firstT,
    const float* __restrict__ Wout, const float* __restrict__ bout,
    const float* __restrict__ scale, const float* __restrict__ xrw,
    float* __restrict__ out) {
  __shared__ float att[16 * LL];
  const int tid = threadIdx.x;
  const int w = tid >> 5, t = tid & 31;
  const int lane = t & 15, half = t >> 4;
  const int g0 = blockIdx.x * 16;   // token-row base over B*N
  const int b  = g0 >> 11;          // tiles never cross batch
  const size_t fb = (size_t)b * RD * LL;

  // phase 1: att[:, l0:l0+16] = q_pe_tile(16xRD) * first(RDxL) column tile
  const int l0 = 16 * w;
  v8f acc = {};
  for (int k = 0; k < RD; k += 4) {
    int kh = k + 2 * half;
    v2f a  = *(const v2f*)&qpe[(size_t)(g0 + lane) * RD + kh];
    v2f bb = *(const v2f*)&firstT[fb + (size_t)(l0 + lane) * RD + kh];
    acc = wmma4(a, bb, acc);
  }
  for (int j = 0; j < 8; ++j)
    att[(j + 8 * half) * LL + l0 + lane] = acc[j];
  __syncthreads();

  // phase 2: out[:, m0:m0+16] = att(16xL) * Wout^T tile + epilogue
  const int m0 = 16 * w;
  v8f acco = {};
  for (int k = 0; k < LL; k += 4) {
    int kh = k + 2 * half;
    v2f a  = *(const v2f*)&att[lane * LL + kh];
    v2f bb = *(const v2f*)&Wout[(size_t)(m0 + lane) * LL + kh];
    acco = wmma4(a, bb, acco);
  }
  for (int j = 0; j < 8; ++j) {
    int row = g0 + j + 8 * half;
    int col = m0 + lane;
    out[(size_t)row * LL + col] =
        acco[j] + bout[col] + scale[row] * xrw[(size_t)row * LL + col];
  }
}

// ---------------------------------------------------------------------------
// host launcher
// inputs (setup_inputs order): x, init_pe, log_deg, deg_coef, Wx, Wq, Wk,
//                              Wout, bout, Wrpe, Wrx, Wrpe2
// ---------------------------------------------------------------------------
extern "C" void kernel_launch(void* const* d_in, const int* in_sizes, int n_in,
                              void* d_out, int out_size, void* d_ws,
                              size_t ws_size, hipStream_t stream) {
  (void)in_sizes; (void)n_in; (void)out_size; (void)ws_size;
  const float* x        = (const float*)d_in[0];
  const float* pe       = (const float*)d_in[1];
  const float* log_deg  = (const float*)d_in[2];
  const float* deg_coef = (const float*)d_in[3];
  const float* Wx       = (const float*)d_in[4];
  const float* Wq       = (const float*)d_in[5];
  const float* Wk       = (const float*)d_in[6];
  const float* Wout     = (const float*)d_in[7];
  const float* bout     = (const float*)d_in[8];
  const float* Wrpe     = (const float*)d_in[9];
  const float* Wrx      = (const float*)d_in[10];
  const float* Wrpe2    = (const float*)d_in[11];
  float* out = (float*)d_out;

  // workspace layout (floats): ~138.6 MB total
  float* ws     = (float*)d_ws;
  float* xhT    = ws;                    // [B][L][N]   (transposed)
  float* xrw    = ws + BNL;              // [B*N, L]
  float* qpe    = ws + 2 * BNL;          // [B*N*R, DPE] == [B*N, RD]
  float* kpeT   = ws + 3 * BNL;          // [B][RD][N]  (transposed)
  float* scl    = ws + 4 * BNL;          // [B*N]
  float* firstT = scl + BN;              // [B][L][RD]  (transposed)

  mix_gemm_kernel<<<BN / 16, 512, 0, stream>>>(x, log_deg, deg_coef, Wx, Wrx,
                                               xhT, xrw);
  pe_gemm_kernel<<<(BN * RR) / 128, 256, 0, stream>>>(pe, Wq, Wk, qpe, kpeT);
  scale_kernel<<<BN / 8, 256, 0, stream>>>(pe, Wrpe, Wrpe2, scl);
  first_kernel<<<dim3(RD / 16, BB), 512, 0, stream>>>(kpeT, xhT, firstT);
  attn_out_kernel<<<BN / 16, 512, 0, stream>>>(qpe, firstT, Wout, bout, scl,
                                               xrw, out);
}